// Graph_Conv_66434554134762
// MI455X (gfx1250) — compile-verified
//
#include <hip/hip_runtime.h>
#include <math.h>

typedef float v2f __attribute__((ext_vector_type(2)));
typedef float v8f __attribute__((ext_vector_type(8)));

// -------------------------------------------------------------------------
// K1: chan_data prep.
// chan_pad[B,64]   : [chan_feats(20) | emb_chan(32) | 0-pad(12)]
// xb/xn[B,128]     : attention inputs; cols 0..51 = chan_data, cols 52..115
//                    filled later by the h-GEMM epilogue, cols 116..127 = 0.
// -------------------------------------------------------------------------
__global__ void chan_prep_kernel(const float* __restrict__ chan_feats,
                                 const int*   __restrict__ chan_id,
                                 const float* __restrict__ emb_chan,
                                 float* __restrict__ chan_pad,
                                 float* __restrict__ xb,
                                 float* __restrict__ xn,
                                 int B) {
  int t = blockIdx.x * blockDim.x + threadIdx.x;
  if (t >= B * 64) return;
  int b = t >> 6, d = t & 63;
  float v = 0.0f;
  if (d < 20)       v = chan_feats[b * 20 + d];
  else if (d < 52)  v = emb_chan[chan_id[b] * 32 + (d - 20)];
  chan_pad[b * 64 + d] = v;
  if (d < 52) {
    xb[b * 128 + d] = v;
    xn[b * 128 + d] = v;
  } else { // d = 52..63 -> zero pad cols 116..127
    xb[b * 128 + 64 + d] = 0.0f;
    xn[b * 128 + 64 + d] = 0.0f;
  }
}

// -------------------------------------------------------------------------
// K2: neighbor gather + mean. One block per channel b, lane = feature dim.
// Lane->segment mapping keeps every embedding-row read coalesced (64-80B
// bursts); all tables are L2-resident (192MB L2 >> 17MB of tables).
// agg[B,80] zero padded (cols 77..79).
// -------------------------------------------------------------------------
__global__ void agg_kernel(const int*   __restrict__ nei,        // [B,64]
                           const float* __restrict__ dev_dense,  // [N,20]
                           const float* __restrict__ dev_extra,  // [N,1]
                           const int*   __restrict__ carrier,
                           const int*   __restrict__ language,
                           const int*   __restrict__ brand,
                           const int*   __restrict__ osidx,
                           const float* __restrict__ e_car,      // [*,16]
                           const float* __restrict__ e_lang,     // [*,16]
                           const float* __restrict__ e_brand,    // [*,16]
                           const float* __restrict__ e_os,       // [*,8]
                           float* __restrict__ agg) {             // [B,80]
  int b = blockIdx.x;
  int d = threadIdx.x;              // blockDim = 96, dims 0..76 active
  const int* neib = nei + b * 64;
  float acc = 0.0f;
  for (int k = 0; k < 64; ++k) {
    int dev = neib[k];              // same addr across lanes -> broadcast
    float v = 0.0f;
    if (d < 20)        v = dev_dense[dev * 20 + d];
    else if (d == 20)  v = dev_extra[dev];
    else if (d < 37)   v = e_car [carrier [dev] * 16 + (d - 21)];
    else if (d < 53)   v = e_lang[language[dev] * 16 + (d - 37)];
    else if (d < 69)   v = e_brand[brand  [dev] * 16 + (d - 53)];
    else if (d < 77)   v = e_os  [osidx  [dev] *  8 + (d - 69)];
    acc += v;
  }
  if (d < 77)      agg[b * 80 + d] = acc * (1.0f / 64.0f);
  else if (d < 80) agg[b * 80 + d] = 0.0f;   // zero pad
}

// -------------------------------------------------------------------------
// K3: fp32 WMMA GEMM + bias + ReLU:  out = relu(A[B,lda] @ W[KREAL,N] + b)
// One 16x16 output tile per wave, K consumed 4 at a time with
// v_wmma_f32_16x16x4_f32. Guarded B-loads avoid 0*NaN from padded K.
// Optional second destination (scatter h into attention input X).
// -------------------------------------------------------------------------
template <int KSTEPS, int KREAL, int N>
__global__ void wmma_gemm_relu_kernel(const float* __restrict__ A, int lda,
                                      const float* __restrict__ W,
                                      const float* __restrict__ bias,
                                      float* __restrict__ out,  int ldo,  int ocol,
                                      float* __restrict__ out2, int ldo2, int o2col,
                                      int ntiles_m) {
  constexpr int NT_N = N / 16;
  int wave = threadIdx.x >> 5;
  int lane = threadIdx.x & 31;
  int tile = blockIdx.x * (blockDim.x >> 5) + wave;
  if (tile >= ntiles_m * NT_N) return;     // wave-uniform -> EXEC stays all-1s
  int tm = tile / NT_N, tn = tile % NT_N;
  int m    = lane & 15;
  int half = lane >> 4;
  int col  = tn * 16 + m;
  const float* arow = A + (long)(tm * 16 + m) * lda;

  v8f c = {0.f, 0.f, 0.f, 0.f, 0.f, 0.f, 0.f, 0.f};
#pragma unroll
  for (int kk = 0; kk < KSTEPS; ++kk) {
    int k0 = kk * 4 + 2 * half;            // fp32 A 16x4 layout: lanes16-31 K=2,3
    v2f a = *(const v2f*)(arow + k0);      // 8B aligned (lda,k0 even)
    v2f bf;
    bf.x = (k0     < KREAL) ? W[(long)k0 * N + col]       : 0.0f;
    bf.y = (k0 + 1 < KREAL) ? W[(long)(k0 + 1) * N + col] : 0.0f;
    c = __builtin_amdgcn_wmma_f32_16x16x4_f32(false, a, false, bf,
                                              (short)0, c, false, false);
  }
  float bv = bias[col];
#pragma unroll
  for (int r = 0; r < 8; ++r) {            // VGPR r: M = r (lanes 0-15), r+8 (16-31)
    int row = tm * 16 + r + 8 * half;
    float v = c[r] + bv;
    v = v > 0.0f ? v : 0.0f;
    out[(long)row * ldo + ocol + col] = v;
    if (out2) out2[(long)row * ldo2 + o2col + col] = v;
  }
}

// -------------------------------------------------------------------------
// K4: attention score.  z = relu(X[B,128] @ W_attn[116,48] + b_attn);
// score = z @ W_attn2 + b_attn2.  Block = 3 waves = 3 N-tiles of one
// 16-row strip; z staged through LDS, 16 threads do the final 48-dot.
// -------------------------------------------------------------------------
__global__ void attn_score_kernel(const float* __restrict__ X,       // [B,128]
                                  const float* __restrict__ W_attn,  // [116,48]
                                  const float* __restrict__ b_attn,  // [48]
                                  const float* __restrict__ W_attn2, // [48]
                                  const float* __restrict__ b_attn2, // [1]
                                  float* __restrict__ scores) {      // [B]
  __shared__ float zsh[16][48];
  int wave = threadIdx.x >> 5;   // 0..2 -> N tile
  int lane = threadIdx.x & 31;
  int tm   = blockIdx.x;
  int m    = lane & 15;
  int half = lane >> 4;
  int col  = wave * 16 + m;
  const float* arow = X + (long)(tm * 16 + m) * 128;

  v8f c = {0.f, 0.f, 0.f, 0.f, 0.f, 0.f, 0.f, 0.f};
#pragma unroll
  for (int kk = 0; kk < 29; ++kk) {        // 29*4 = 116 = real K
    int k0 = kk * 4 + 2 * half;
    v2f a = *(const v2f*)(arow + k0);
    v2f bf;
    bf.x = (k0     < 116) ? W_attn[k0 * 48 + col]       : 0.0f;
    bf.y = (k0 + 1 < 116) ? W_attn[(k0 + 1) * 48 + col] : 0.0f;
    c = __builtin_amdgcn_wmma_f32_16x16x4_f32(false, a, false, bf,
                                              (short)0, c, false, false);
  }
  float bv = b_attn[col];
#pragma unroll
  for (int r = 0; r < 8; ++r) {
    float v = c[r] + bv;
    zsh[r + 8 * half][col] = v > 0.0f ? v : 0.0f;
  }
  __syncthreads();
  if (threadIdx.x < 16) {
    float acc = b_attn2[0];
#pragma unroll
    for (int j = 0; j < 48; ++j) acc += zsh[threadIdx.x][j] * W_attn2[j];
    scores[tm * 16 + threadIdx.x] = acc;
  }
}

// -------------------------------------------------------------------------
// K5: 2-way softmax over scores, blend h_bots/h_normal into d_out[:,64:128].
// -------------------------------------------------------------------------
__global__ void combine_kernel(const float* __restrict__ scores,  // [2,B]
                               const float* __restrict__ hb,      // [B,64]
                               const float* __restrict__ hn,      // [B,64]
                               float* __restrict__ out,           // [B,128]
                               int B) {
  int t = blockIdx.x * blockDim.x + threadIdx.x;
  if (t >= B * 64) return;
  int b = t >> 6, n = t & 63;
  float sb = scores[b], sn = scores[B + b];
  float mx = fmaxf(sb, sn);
  float eb = expf(sb - mx), en = expf(sn - mx);
  float cb = eb / (eb + en);
  out[b * 128 + 64 + n] = hb[b * 64 + n] * cb + hn[b * 64 + n] * (1.0f - cb);
}

// -------------------------------------------------------------------------
extern "C" void kernel_launch(void* const* d_in, const int* in_sizes, int n_in,
                              void* d_out, int out_size, void* d_ws, size_t ws_size,
                              hipStream_t stream) {
  const float* chan_feats = (const float*)d_in[0];
  const int*   chan_id    = (const int*)  d_in[1];
  const float* dev_dense  = (const float*)d_in[2];
  const float* dev_extra  = (const float*)d_in[3];
  const int*   carrier    = (const int*)  d_in[4];
  const int*   language   = (const int*)  d_in[5];
  const int*   brand      = (const int*)  d_in[6];
  const int*   osidx      = (const int*)  d_in[7];
  const int*   nei_bots   = (const int*)  d_in[8];
  const int*   nei_norm   = (const int*)  d_in[9];
  const float* e_car      = (const float*)d_in[10];
  const float* e_lang     = (const float*)d_in[11];
  const float* e_brand    = (const float*)d_in[12];
  const float* e_os       = (const float*)d_in[13];
  const float* e_chan     = (const float*)d_in[14];
  const float* W_agg_b    = (const float*)d_in[15];
  const float* b_agg_b    = (const float*)d_in[16];
  const float* W_agg_n    = (const float*)d_in[17];
  const float* b_agg_n    = (const float*)d_in[18];
  const float* W_self     = (const float*)d_in[19];
  const float* b_self     = (const float*)d_in[20];
  const float* W_attn     = (const float*)d_in[21];
  const float* b_attn     = (const float*)d_in[22];
  const float* W_attn2    = (const float*)d_in[23];
  const float* b_attn2    = (const float*)d_in[24];

  const int B = in_sizes[0] / 20;   // 8192
  float* out = (float*)d_out;

  // Workspace carve-up (~20MB, all L2-resident)
  float* ws       = (float*)d_ws;
  float* aggB     = ws;  ws += (size_t)B * 80;
  float* aggN     = ws;  ws += (size_t)B * 80;
  float* chan_pad = ws;  ws += (size_t)B * 64;
  float* xb       = ws;  ws += (size_t)B * 128;
  float* xn       = ws;  ws += (size_t)B * 128;
  float* hb       = ws;  ws += (size_t)B * 64;
  float* hn       = ws;  ws += (size_t)B * 64;
  float* scores   = ws;  ws += (size_t)2 * B;
  (void)n_in; (void)out_size; (void)ws_size;

  const int ntm = B / 16;                    // 512 row tiles
  const int gemm_blocks = (ntm * 4 + 7) / 8; // 8 waves (tiles) per 256-thr block

  // K1: channel features + channel-id embedding (also seeds attention X)
  chan_prep_kernel<<<(B * 64 + 255) / 256, 256, 0, stream>>>(
      chan_feats, chan_id, e_chan, chan_pad, xb, xn, B);

  // K2: neighbor gather + mean, both neighbor sets
  agg_kernel<<<B, 96, 0, stream>>>(nei_bots, dev_dense, dev_extra, carrier,
      language, brand, osidx, e_car, e_lang, e_brand, e_os, aggB);
  agg_kernel<<<B, 96, 0, stream>>>(nei_norm, dev_dense, dev_extra, carrier,
      language, brand, osidx, e_car, e_lang, e_brand, e_os, aggN);

  // K3: h = relu(agg @ W_agg + b) (fp32 WMMA); epilogue scatters into X col 52
  wmma_gemm_relu_kernel<20, 77, 64><<<gemm_blocks, 256, 0, stream>>>(
      aggB, 80, W_agg_b, b_agg_b, hb, 64, 0, xb, 128, 52, ntm);
  wmma_gemm_relu_kernel<20, 77, 64><<<gemm_blocks, 256, 0, stream>>>(
      aggN, 80, W_agg_n, b_agg_n, hn, 64, 0, xn, 128, 52, ntm);

  // self_h = relu(chan_data @ W_self + b) -> d_out[:, 0:64]
  wmma_gemm_relu_kernel<13, 52, 64><<<gemm_blocks, 256, 0, stream>>>(
      chan_pad, 64, W_self, b_self, out, 128, 0, (float*)nullptr, 0, 0, ntm);

  // K4: attention scores for both branches
  attn_score_kernel<<<ntm, 96, 0, stream>>>(xb, W_attn, b_attn, W_attn2,
                                            b_attn2, scores);
  attn_score_kernel<<<ntm, 96, 0, stream>>>(xn, W_attn, b_attn, W_attn2,
                                            b_attn2, scores + B);

  // K5: softmax blend -> d_out[:, 64:128]
  combine_kernel<<<(B * 64 + 255) / 256, 256, 0, stream>>>(scores, hb, hn, out, B);
}